// GCMCGraphConv_223338299478
// MI455X (gfx1250) — compile-verified
//
#include <hip/hip_runtime.h>

// GCMC graph conv for MI455X (gfx1250, wave32).
//   K1: zero d_out (harness poisons it; segment_sum needs 0-init for atomics)
//   K2: h = (A @ W) * cj  via V_WMMA_F32_16X16X4_F32, W staged in LDS in a
//       K-pair-interleaved float2 layout so every B fragment is ONE ds_load_b64
//       landing in an even-aligned VGPR pair (no repack movs before v_wmma).
//   K3: out[dst] += h[src] * ci[dst]  (coalesced gather + global_atomic_add_f32;
//       h and out are L2-resident: 25.6 MB each vs 192 MB L2)

typedef __attribute__((ext_vector_type(2))) float v2f;
typedef __attribute__((ext_vector_type(8))) float v8f;

#define D_IN  128
#define D_OUT 64

__global__ void gcmc_zero_kernel(float4* __restrict__ out, int n4) {
    int i = blockIdx.x * blockDim.x + threadIdx.x;
    if (i < n4) out[i] = make_float4(0.f, 0.f, 0.f, 0.f);
}

// One wave computes a 16x64 tile of h. 8 waves / 256-thread block.
__global__ __launch_bounds__(256)
void gcmc_gemm_wmma_kernel(const float* __restrict__ A,   // [N, 128]
                           const float* __restrict__ W,   // [128, 64]
                           const float* __restrict__ cj,  // [N]
                           float* __restrict__ H,         // [N, 64] (ws)
                           int nTiles) {
    // K-pair interleaved weight: sW2[kk*64 + n] = {W[2kk][n], W[2kk+1][n]}
    __shared__ float2 sW2[(D_IN / 2) * D_OUT];  // 32 KB of 320 KB/WGP

    {
        int tid = threadIdx.x;
        #pragma unroll
        for (int i = 0; i < ((D_IN / 2) * D_OUT) / 256; ++i) {
            int p  = tid + i * 256;
            int kk = p >> 6;          // pair index 0..63
            int n  = p & 63;          // column (consecutive across lanes: coalesced)
            float2 w;
            w.x = W[(2 * kk)     * D_OUT + n];
            w.y = W[(2 * kk + 1) * D_OUT + n];
            sW2[p] = w;               // ds_store_b64
        }
    }
    __syncthreads();

    const int wave = threadIdx.x >> 5;
    const int lane = threadIdx.x & 31;
    const int tile = blockIdx.x * (blockDim.x >> 5) + wave;
    if (tile >= nTiles) return;          // wave-uniform: EXEC stays all-1s for WMMA

    const int half = lane >> 4;          // 0: lanes 0-15, 1: lanes 16-31
    const int lrow = lane & 15;

    // ISA 32-bit A-matrix 16x4 layout: lanes 0-15 carry K=k,k+1; lanes 16-31 K=k+2,k+3.
    const float* Arow = A + (long long)(tile * 16 + lrow) * D_IN;

    v8f c0 = {}, c1 = {}, c2 = {}, c3 = {};

    for (int k = 0; k < D_IN; k += 4) {
        const int kA = k + 2 * half;        // even
        v2f a = *(const v2f*)(Arow + kA);   // one global_load_b64 per lane per step

        // kk = kA/2; four B fragments = four ds_load_b64 at immediate offsets
        const float2* sRow = sW2 + ((k >> 1) + half) * D_OUT + lrow;
        v2f b0 = *(const v2f*)(sRow +  0);
        v2f b1 = *(const v2f*)(sRow + 16);
        v2f b2 = *(const v2f*)(sRow + 32);
        v2f b3 = *(const v2f*)(sRow + 48);

        c0 = __builtin_amdgcn_wmma_f32_16x16x4_f32(false, a, false, b0, (short)0, c0, false, false);
        c1 = __builtin_amdgcn_wmma_f32_16x16x4_f32(false, a, false, b1, (short)0, c1, false, false);
        c2 = __builtin_amdgcn_wmma_f32_16x16x4_f32(false, a, false, b2, (short)0, c2, false, false);
        c3 = __builtin_amdgcn_wmma_f32_16x16x4_f32(false, a, false, b3, (short)0, c3, false, false);
    }

    // C/D layout: VGPR r, lanes 0-15 -> M=r, N=lane; lanes 16-31 -> M=r+8, N=lane-16.
    // Fuse the cj (src-norm) scale into the epilogue.
    #pragma unroll
    for (int r = 0; r < 8; ++r) {
        const int   node = tile * 16 + r + 8 * half;
        const float s    = cj[node];
        float* Hrow = H + (long long)node * D_OUT;
        Hrow[ 0 + lrow] = c0[r] * s;
        Hrow[16 + lrow] = c1[r] * s;
        Hrow[32 + lrow] = c2[r] * s;
        Hrow[48 + lrow] = c3[r] * s;
    }
}

// 32 threads per edge, float2 per thread -> 64 cols. ci folded per-message:
// sum_e h[src_e]*ci[d] == ci[d]*sum_e h[src_e].
__global__ void gcmc_scatter_kernel(const float* __restrict__ H,
                                    const float* __restrict__ ci,
                                    const int*   __restrict__ src,
                                    const int*   __restrict__ dst,
                                    float*       __restrict__ out,
                                    int nEdges) {
    long long idx   = (long long)blockIdx.x * blockDim.x + threadIdx.x;
    long long nwork = (long long)nEdges * 32;
    if (idx >= nwork) return;
    const int e = (int)(idx >> 5);
    const int c = (int)(idx & 31) * 2;
    const int s = src[e];
    const int d = dst[e];
    const float  cd = ci[d];
    const float2 hv = *(const float2*)(H + (long long)s * D_OUT + c);
    float* o = out + (long long)d * D_OUT + c;
    atomicAdd(o,     hv.x * cd);   // global_atomic_add_f32, no return
    atomicAdd(o + 1, hv.y * cd);
}

extern "C" void kernel_launch(void* const* d_in, const int* in_sizes, int n_in,
                              void* d_out, int out_size, void* d_ws, size_t ws_size,
                              hipStream_t stream) {
    const float* input_feat = (const float*)d_in[0];   // [N,128]
    const float* weight     = (const float*)d_in[1];   // [128,64]
    const float* cj         = (const float*)d_in[2];   // [N,1]
    const float* ci         = (const float*)d_in[3];   // [N,1]
    const int*   src_idx    = (const int*)d_in[4];     // [E]
    const int*   dst_idx    = (const int*)d_in[5];     // [E]
    float*       out        = (float*)d_out;           // [N,64]
    float*       H          = (float*)d_ws;            // [N,64] scratch (25.6 MB)

    const int N = in_sizes[2];        // 100000
    const int E = in_sizes[4];        // 1600000

    // K1: zero output for atomic accumulation
    {
        const int n4 = out_size / 4;
        gcmc_zero_kernel<<<(n4 + 255) / 256, 256, 0, stream>>>((float4*)out, n4);
    }

    // K2: WMMA GEMM + cj scale (N divisible by 16 for this problem)
    {
        const int nTiles = N / 16;                     // 6250
        const int blocks = (nTiles + 7) / 8;           // 8 waves/block
        gcmc_gemm_wmma_kernel<<<blocks, 256, 0, stream>>>(input_feat, weight, cj, H, nTiles);
    }

    // K3: edge scatter with ci fold
    {
        const long long nwork  = (long long)E * 32;
        const int       blocks = (int)((nwork + 255) / 256);
        gcmc_scatter_kernel<<<blocks, 256, 0, stream>>>(H, ci, src_idx, dst_idx, out, E);
    }
}